// RNN_57449482551561
// MI455X (gfx1250) — compile-verified
//
#include <hip/hip_runtime.h>
#include <hip/hip_bf16.h>

typedef __attribute__((ext_vector_type(2))) float v2f;
typedef __attribute__((ext_vector_type(8))) float v8f;

#define T_STEPS   512
#define BATCH     64
#define HID       1024
#define NCLS      32000
#define W1_LD     (6 * HID)            // w1 is (H, 6H) row-major
#define TH_STRIDE (T_STEPS * HID)      // per-batch stride inside 'out' region
#define RNN_BLOCKS 16                  // persistent kernel block count (co-resident)

// D = A(16x4) * B(4x16) + C  in fp32 (V_WMMA_F32_16X16X4_F32)
__device__ __forceinline__ v8f wmma_f32(v2f a, v2f b, v8f c) {
    return __builtin_amdgcn_wmma_f32_16x16x4_f32(
        /*neg_a=*/false, a, /*neg_b=*/false, b,
        /*c_mod=*/(short)0, c, /*reuse_a=*/false, /*reuse_b=*/false);
}

// ---------------- barrier init (ws is poisoned by harness; reset every call) --
__global__ void k_init(unsigned* bar) {
    bar[0] = 0u;   // arrival count
    bar[1] = 0u;   // generation
}

// ---------------- device-wide sense-reversal barrier ------------------------
__device__ __forceinline__ unsigned grid_sync(unsigned* bar, unsigned gen,
                                              unsigned ngroups) {
    __threadfence();                 // release my stores to device scope
    __syncthreads();
    if (threadIdx.x == 0) {
        unsigned prev = atomicAdd(&bar[0], 1u);
        if (prev == ngroups - 1u) {
            atomicExch(&bar[0], 0u);         // reset for next round
            atomicAdd(&bar[1], 1u);          // open the gate
        } else {
            while (atomicAdd(&bar[1], 0u) <= gen)
                __builtin_amdgcn_s_sleep(1);
        }
    }
    __syncthreads();
    __threadfence();                 // acquire: invalidate stale cache lines
    return gen + 1u;
}

// ---------------- kernel 1: xin = emb[x] @ Wx + b ---------------------------
// M = T*B = 32768 rows, N = K = 1024.  Wave owns one 16-row M tile and a
// 32-column N pair.  Four independent WMMA chains (2 N-tiles x 2 K-phases)
// give dependence distance 4 on the D->C accumulate.
__global__ __launch_bounds__(256) void k_xin(const int* __restrict__ x,
                                             const float* __restrict__ emb,
                                             const float* __restrict__ w1,
                                             const float* __restrict__ bias,
                                             float* __restrict__ xin) {
    const int lane = threadIdx.x & 31;
    const int warp = threadIdx.x >> 5;
    const int wid  = blockIdx.x * 8 + warp;     // 0 .. 65535
    const int m0   = (wid >> 5) * 16;           // 2048 M tiles
    const int n0   = (wid & 31) * 32;           // 32 N pairs
    const int row  = lane & 15;                 // M (A) / N (B) within tile
    const int kb   = (lane >> 4) * 2;           // K sub-offset per ISA layout

    const int tok = x[m0 + row];                        // gathered embedding row
    const float* __restrict__ arow = emb + (size_t)tok * HID;

    v8f acc0a = {}, acc1a = {}, acc0b = {}, acc1b = {};
#pragma unroll 2
    for (int k = 0; k < HID; k += 8) {
        {   // K phase a: k .. k+3
            v2f a = *(const v2f*)(arow + k + kb);
            const float* wr0 = w1 + (size_t)(k + kb) * W1_LD;
            const float* wr1 = w1 + (size_t)(k + kb + 1) * W1_LD;
            v2f b0 = { wr0[n0 + row],      wr1[n0 + row]      };
            v2f b1 = { wr0[n0 + 16 + row], wr1[n0 + 16 + row] };
            acc0a = wmma_f32(a, b0, acc0a);
            acc1a = wmma_f32(a, b1, acc1a);
        }
        {   // K phase b: k+4 .. k+7
            v2f a = *(const v2f*)(arow + k + 4 + kb);
            const float* wr0 = w1 + (size_t)(k + 4 + kb) * W1_LD;
            const float* wr1 = w1 + (size_t)(k + 4 + kb + 1) * W1_LD;
            v2f b0 = { wr0[n0 + row],      wr1[n0 + row]      };
            v2f b1 = { wr0[n0 + 16 + row], wr1[n0 + 16 + row] };
            acc0b = wmma_f32(a, b0, acc0b);
            acc1b = wmma_f32(a, b1, acc1b);
        }
    }
    const v8f acc0 = acc0a + acc0b;
    const v8f acc1 = acc1a + acc1b;

    const int nA = n0 + row, nB = n0 + 16 + row;
    const float biasA = bias[nA], biasB = bias[nB];
#pragma unroll
    for (int g = 0; g < 8; ++g) {
        const int m = m0 + g + ((lane >> 4) ? 8 : 0);   // C/D VGPR layout
        xin[(size_t)m * HID + nA] = acc0[g] + biasA;
        xin[(size_t)m * HID + nB] = acc1[g] + biasB;
    }
}

// ---------------- kernel 2: h0 = tanh(xin[0]) -> out slot 0 -----------------
__global__ void k_h0(const float* __restrict__ xin, float* __restrict__ outp) {
    const int i = blockIdx.x * blockDim.x + threadIdx.x;    // 64*1024
    const int b = i >> 10, n = i & 1023;
    outp[(size_t)b * TH_STRIDE + n] = tanhf(xin[i]);
}

// ---------------- kernel 3: persistent recurrence ---------------------------
// 16 blocks x 8 waves = 128 waves; wave owns a 16(M) x 32(N) slab of h.
// Hidden states live directly in the 'out' region of d_out: out[b, t*H + n].
__global__ __launch_bounds__(256) void k_rnn(const float* __restrict__ w1,
                                             const float* __restrict__ xin,
                                             float* __restrict__ outp,
                                             unsigned* __restrict__ bar) {
    const int lane = threadIdx.x & 31;
    const int warp = threadIdx.x >> 5;
    const int wid  = blockIdx.x * 8 + warp;     // 0 .. 127
    const int m0   = (wid & 3) * 16;            // 4 M tiles (batch = 64)
    const int n0   = (wid >> 2) * 32;           // 32 N pairs (H = 1024)
    const int row  = lane & 15;
    const int kb   = (lane >> 4) * 2;

    const float* __restrict__ Wh = w1 + HID;    // Wh[k][n] = w1[k*6H + H + n]
    unsigned gen = 0;

    for (int t = 1; t < T_STEPS; ++t) {
        const float* __restrict__ xt   = xin + (size_t)t * BATCH * HID;
        float* __restrict__       hdst = outp + (size_t)t * HID;
#pragma unroll 1
        for (int it = 0; it < 2; ++it) {        // HID_NUM = 2
            const float* __restrict__ hsrc =
                outp + (size_t)(it == 0 ? (t - 1) : t) * HID;
            const float* __restrict__ arow =
                hsrc + (size_t)(m0 + row) * TH_STRIDE;

            v8f acc0a = {}, acc1a = {}, acc0b = {}, acc1b = {};
#pragma unroll 2
            for (int k = 0; k < HID; k += 8) {
                {
                    v2f a = *(const v2f*)(arow + k + kb);
                    const float* wr0 = Wh + (size_t)(k + kb) * W1_LD;
                    const float* wr1 = Wh + (size_t)(k + kb + 1) * W1_LD;
                    v2f b0 = { wr0[n0 + row],      wr1[n0 + row]      };
                    v2f b1 = { wr0[n0 + 16 + row], wr1[n0 + 16 + row] };
                    acc0a = wmma_f32(a, b0, acc0a);
                    acc1a = wmma_f32(a, b1, acc1a);
                }
                {
                    v2f a = *(const v2f*)(arow + k + 4 + kb);
                    const float* wr0 = Wh + (size_t)(k + 4 + kb) * W1_LD;
                    const float* wr1 = Wh + (size_t)(k + 4 + kb + 1) * W1_LD;
                    v2f b0 = { wr0[n0 + row],      wr1[n0 + row]      };
                    v2f b1 = { wr0[n0 + 16 + row], wr1[n0 + 16 + row] };
                    acc0b = wmma_f32(a, b0, acc0b);
                    acc1b = wmma_f32(a, b1, acc1b);
                }
            }
            const v8f acc0 = acc0a + acc0b;
            const v8f acc1 = acc1a + acc1b;

            const int nA = n0 + row, nB = n0 + 16 + row;
#pragma unroll
            for (int g = 0; g < 8; ++g) {
                const int m = m0 + g + ((lane >> 4) ? 8 : 0);
                hdst[(size_t)m * TH_STRIDE + nA] =
                    tanhf(acc0[g] + xt[(size_t)m * HID + nA]);
                hdst[(size_t)m * TH_STRIDE + nB] =
                    tanhf(acc1[g] + xt[(size_t)m * HID + nB]);
            }
            gen = grid_sync(bar, gen, RNN_BLOCKS);
        }
    }
}

// ---------------- kernel 4: y = h_last @ fc_w^T + fc_b ----------------------
// M = 64, N = 32000, K = 1024.  HBM-bound (streams 131 MB of fc_w).
__global__ __launch_bounds__(256) void k_fc(const float* __restrict__ outp,
                                            const float* __restrict__ fc_w,
                                            const float* __restrict__ fc_b,
                                            float* __restrict__ y) {
    const int lane = threadIdx.x & 31;
    const int warp = threadIdx.x >> 5;
    const int wid  = blockIdx.x * 8 + warp;     // 0 .. 3999
    const int m0   = (wid & 3) * 16;
    const int n0   = (wid >> 2) * 32;           // 1000 N pairs cover C=32000
    const int row  = lane & 15;
    const int kb   = (lane >> 4) * 2;

    const float* __restrict__ hlast = outp + (size_t)(T_STEPS - 1) * HID;
    const float* __restrict__ arow  = hlast + (size_t)(m0 + row) * TH_STRIDE;
    // B[k][n] = fc_w[n][k] -> contiguous float2 per lane along K (streaming)
    const float* __restrict__ br0 = fc_w + (size_t)(n0 + row) * HID;
    const float* __restrict__ br1 = fc_w + (size_t)(n0 + 16 + row) * HID;
    __builtin_prefetch(br0, 0, 0);              // global_prefetch_b8
    __builtin_prefetch(br1, 0, 0);

    v8f acc0a = {}, acc1a = {}, acc0b = {}, acc1b = {};
#pragma unroll 2
    for (int k = 0; k < HID; k += 8) {
        {
            v2f a  = *(const v2f*)(arow + k + kb);
            v2f b0 = *(const v2f*)(br0 + k + kb);
            v2f b1 = *(const v2f*)(br1 + k + kb);
            acc0a = wmma_f32(a, b0, acc0a);
            acc1a = wmma_f32(a, b1, acc1a);
        }
        {
            v2f a  = *(const v2f*)(arow + k + 4 + kb);
            v2f b0 = *(const v2f*)(br0 + k + 4 + kb);
            v2f b1 = *(const v2f*)(br1 + k + 4 + kb);
            acc0b = wmma_f32(a, b0, acc0b);
            acc1b = wmma_f32(a, b1, acc1b);
        }
    }
    const v8f acc0 = acc0a + acc0b;
    const v8f acc1 = acc1a + acc1b;

    const int nA = n0 + row, nB = n0 + 16 + row;
    const float biasA = fc_b[nA], biasB = fc_b[nB];
#pragma unroll
    for (int g = 0; g < 8; ++g) {
        const int m = m0 + g + ((lane >> 4) ? 8 : 0);
        y[(size_t)m * NCLS + nA] = acc0[g] + biasA;
        y[(size_t)m * NCLS + nB] = acc1[g] + biasB;
    }
}

extern "C" void kernel_launch(void* const* d_in, const int* in_sizes, int n_in,
                              void* d_out, int out_size, void* d_ws, size_t ws_size,
                              hipStream_t stream) {
    const int*   x    = (const int*)d_in[0];
    const float* emb  = (const float*)d_in[1];
    const float* w1   = (const float*)d_in[2];
    const float* b1   = (const float*)d_in[3];
    const float* fc_w = (const float*)d_in[4];
    const float* fc_b = (const float*)d_in[5];

    float* y    = (float*)d_out;                         // (B, C)
    float* outp = y + (size_t)BATCH * NCLS;              // (B, T*H) hidden states
    unsigned* bar = (unsigned*)d_ws;                     // 2 counters
    float* xin = (float*)((char*)d_ws + 256);            // T*B*H f32 scratch

    k_init<<<1, 1, 0, stream>>>(bar);
    k_xin<<<8192, 256, 0, stream>>>(x, emb, w1, b1, xin);
    k_h0 <<<(BATCH * HID) / 256, 256, 0, stream>>>(xin, outp);
    k_rnn<<<RNN_BLOCKS, 256, 0, stream>>>(w1, xin, outp, bar);
    k_fc <<<500, 256, 0, stream>>>(outp, fc_w, fc_b, y);
}